// DFMAtt_7095285973227
// MI455X (gfx1250) — compile-verified
//
#include <hip/hip_runtime.h>
#include <hip/hip_bf16.h>
#include <math.h>

// Problem constants (match reference)
#define BB   8
#define CIN  256
#define COUT 256
#define KK   4
#define HH   128
#define WW   128
#define HWp  (HH * WW)          // 16384 pixels per image
#define NP   (BB * HWp)         // 131072 total pixels

typedef float v2f __attribute__((ext_vector_type(2)));
typedef float v8f __attribute__((ext_vector_type(8)));

// ---------------------------------------------------------------------------
// Kernel 1: proj GEMM (WMMA f32 16x16x4, 2 o-tiles/wave) + flow/logit heads.
//   4 waves/block, each wave owns a 16-pixel strip; x strip staged in LDS.
//   proj written CHANNEL-LAST: proj[p*COUT + o]  (enables contiguous gathers).
//   sx/sy are absolute sampling coords (grid+flow; normalization cancels).
// ---------------------------------------------------------------------------
__global__ __launch_bounds__(128) void dfm_k1_proj(
    const float* __restrict__ x,      // [B,CIN,H,W]
    const float* __restrict__ conv_w, // [COUT,CIN]
    const float* __restrict__ conv_b, // [COUT]
    const float* __restrict__ off_w,  // [K,2,CIN]
    const float* __restrict__ off_b,  // [K,2]
    const float* __restrict__ wgt_w,  // [K,CIN]
    const float* __restrict__ wgt_b,  // [K]
    float* __restrict__ proj,         // [NP,COUT] channel-last
    float* __restrict__ wts,          // [NP,K]
    float* __restrict__ sx,           // [NP,K]
    float* __restrict__ sy)           // [NP,K]
{
    __shared__ float lds[4 * CIN * 16];      // 64 KB: 4 waves x (256 ch x 16 px)
    const int lane = threadIdx.x & 31;
    const int wave = threadIdx.x >> 5;
    const int p0   = blockIdx.x * 64 + wave * 16;   // first pixel of this wave
    const int b    = p0 >> 14;                      // p0 / HW (HW=16384)
    const int prem = p0 & (HWp - 1);
    const float* xb = x + (size_t)b * CIN * HWp;
    float* l = lds + wave * (CIN * 16);

    // Stage x strip: l[c*16 + n] = x[b, c, prem + n]; coalesced 64B rows.
    for (int i = lane; i < CIN * 16; i += 32) {
        int c = i >> 4, n = i & 15;
        l[c * 16 + n] = xb[(size_t)c * HWp + prem + n];
    }
    __syncthreads();

    const int half = lane >> 4;       // 0: K=0,1  1: K=2,3 (per WMMA A/B layout)
    const int mrow = lane & 15;       // A row (out channel) == B col (pixel)
    const int kp   = half * 2;

    // 8 iterations x 2 output-channel tiles; B (LDS) shared by both wmmas.
    for (int ot = 0; ot < 16; ot += 2) {
        const int o0 = ot * 16;
        v8f acc0 = {}, acc1 = {};
        for (int c0 = 0; c0 < CIN; c0 += 4) {
            v2f a0, a1, bm;
            const float* wr0 = conv_w + (size_t)(o0 + mrow) * CIN + c0 + kp;
            const float* wr1 = wr0 + 16 * CIN;
            a0.x = wr0[0]; a0.y = wr0[1];
            a1.x = wr1[0]; a1.y = wr1[1];
            bm.x = l[(c0 + kp) * 16 + mrow];
            bm.y = l[(c0 + kp + 1) * 16 + mrow];
            acc0 = __builtin_amdgcn_wmma_f32_16x16x4_f32(
                false, a0, false, bm, (short)0, acc0, false, false);
            acc1 = __builtin_amdgcn_wmma_f32_16x16x4_f32(
                false, a1, false, bm, (short)0, acc1, false, false);
        }
        // D layout: VGPR r -> (o = o0 + r + 8*half, pixel = p0 + mrow)
        float* dst0 = proj + (size_t)(p0 + mrow) * COUT + o0 + 8 * half;
        const float* cb0 = conv_b + o0 + 8 * half;
        #pragma unroll
        for (int r = 0; r < 8; ++r) dst0[r] = acc0[r] + cb0[r];
        float* dst1 = dst0 + 16;
        const float* cb1 = cb0 + 16;
        #pragma unroll
        for (int r = 0; r < 8; ++r) dst1[r] = acc1[r] + cb1[r];
    }

    // ---- heads: logits(+softmax) on lanes 0-15, flows on lanes 16-31 ----
    const int n = mrow;
    const int p = p0 + n;
    const int hpix = (p & (HWp - 1)) >> 7;   // /W
    const int wpix = p & (WW - 1);

    if (half == 0) {
        float lg[KK];
        #pragma unroll
        for (int k = 0; k < KK; ++k) {
            float s = wgt_b[k];
            const float* wr = wgt_w + k * CIN;
            for (int c = 0; c < CIN; ++c) s += wr[c] * l[c * 16 + n];
            lg[k] = s;
        }
        float mx = fmaxf(fmaxf(lg[0], lg[1]), fmaxf(lg[2], lg[3]));
        float e[KK], sum = 0.f;
        #pragma unroll
        for (int k = 0; k < KK; ++k) { e[k] = __expf(lg[k] - mx); sum += e[k]; }
        float inv = 1.f / sum;
        #pragma unroll
        for (int k = 0; k < KK; ++k) wts[(size_t)p * KK + k] = e[k] * inv;
    } else {
        #pragma unroll
        for (int k = 0; k < KK; ++k) {
            float fx = off_b[k * 2 + 0], fy = off_b[k * 2 + 1];
            const float* wrx = off_w + (size_t)(k * 2 + 0) * CIN;
            const float* wry = off_w + (size_t)(k * 2 + 1) * CIN;
            for (int c = 0; c < CIN; ++c) {
                float xv = l[c * 16 + n];
                fx += wrx[c] * xv;
                fy += wry[c] * xv;
            }
            // absolute sample coords: normalize->unnormalize cancels exactly
            sx[(size_t)p * KK + k] = (float)wpix + fx;
            sy[(size_t)p * KK + k] = (float)hpix + fy;
        }
    }
}

// ---------------------------------------------------------------------------
// Kernel 2: deformable bilinear gather + weighted sum + eval BatchNorm.
//   One wave per pixel; lane owns 8 channels (2x float4 = global_load_b128).
//   Coords/weights loaded as float4 (b128); corner rows get global_prefetch.
//   feat written CHANNEL-LAST for K3's strip staging.
// ---------------------------------------------------------------------------
__global__ __launch_bounds__(256) void dfm_k2_sample_bn(
    const float* __restrict__ proj,   // [NP,COUT] channel-last
    const float* __restrict__ wts,    // [NP,K]
    const float* __restrict__ sx,     // [NP,K]
    const float* __restrict__ sy,     // [NP,K]
    const float* __restrict__ gamma,
    const float* __restrict__ beta,
    const float* __restrict__ rmean,
    const float* __restrict__ rvar,
    float* __restrict__ feat)         // [NP,CIN] channel-last
{
    const int lane = threadIdx.x & 31;
    const int wave = threadIdx.x >> 5;
    const int p = blockIdx.x * 8 + wave;
    const int b = p >> 14;
    const int c = lane * 8;

    // All per-pixel metadata in three b128 loads.
    const float4 sx4 = *(const float4*)(sx  + (size_t)p * KK);
    const float4 sy4 = *(const float4*)(sy  + (size_t)p * KK);
    const float4 wk4 = *(const float4*)(wts + (size_t)p * KK);
    const float xs[KK] = { sx4.x, sx4.y, sx4.z, sx4.w };
    const float ys[KK] = { sy4.x, sy4.y, sy4.z, sy4.w };
    const float ws_[KK] = { wk4.x, wk4.y, wk4.z, wk4.w };

    const float4* self = (const float4*)(proj + (size_t)p * COUT + c);
    float4 a0 = self[0], a1 = self[1];               // feat starts as proj
    const float* pb = proj + (size_t)b * HWp * COUT; // image base (ch-last)

    // CDNA5 prefetch hint for each offset's base corner line (global_prefetch_b8).
    #pragma unroll
    for (int k = 0; k < KK; ++k) {
        const int xi = min(max((int)floorf(xs[k]), 0), WW - 1);
        const int yi = min(max((int)floorf(ys[k]), 0), HH - 1);
        __builtin_prefetch(pb + ((size_t)(yi * WW + xi)) * COUT + c, 0, 1);
    }

    #pragma unroll
    for (int k = 0; k < KK; ++k) {
        const float xk = xs[k], yk = ys[k], wk = ws_[k];
        const float x0 = floorf(xk), y0 = floorf(yk);
        const float wx1 = xk - x0, wy1 = yk - y0;
        const int xi = (int)x0, yi = (int)y0;
        const float cw[4] = { (1.f - wx1) * (1.f - wy1), wx1 * (1.f - wy1),
                              (1.f - wx1) * wy1,         wx1 * wy1 };
        #pragma unroll
        for (int t = 0; t < 4; ++t) {
            const int xc = xi + (t & 1);
            const int yc = yi + (t >> 1);
            const bool valid = (xc >= 0) & (xc < WW) & (yc >= 0) & (yc < HH);
            const float g = wk * cw[t] * (valid ? 1.f : 0.f); // zeros padding
            const int xcc = min(max(xc, 0), WW - 1);
            const int ycc = min(max(yc, 0), HH - 1);
            const float4* src =
                (const float4*)(pb + ((size_t)(ycc * WW + xcc)) * COUT + c);
            float4 v0 = src[0], v1 = src[1];
            a0.x += g * v0.x; a0.y += g * v0.y; a0.z += g * v0.z; a0.w += g * v0.w;
            a1.x += g * v1.x; a1.y += g * v1.y; a1.z += g * v1.z; a1.w += g * v1.w;
        }
    }

    // eval BatchNorm, then store channel-last
    float acc[8] = { a0.x, a0.y, a0.z, a0.w, a1.x, a1.y, a1.z, a1.w };
    float out[8];
    #pragma unroll
    for (int r = 0; r < 8; ++r) {
        const int ch = c + r;
        const float inv = gamma[ch] * rsqrtf(rvar[ch] + 1e-5f);
        out[r] = (acc[r] - rmean[ch]) * inv + beta[ch];
    }
    float4* dst = (float4*)(feat + (size_t)p * CIN + c);
    dst[0] = make_float4(out[0], out[1], out[2], out[3]);
    dst[1] = make_float4(out[4], out[5], out[6], out[7]);
}

// ---------------------------------------------------------------------------
// Kernel 3: output GEMM (WMMA f32 16x16x4, 2 o-tiles/wave):
//   out[b,o,h,w] = out_w^T * feat.  A(o,c) = out_w[c*COUT+o] (coalesced over o);
//   B strip staged in LDS and shared by both accumulators.
// ---------------------------------------------------------------------------
__global__ __launch_bounds__(128) void dfm_k3_out(
    const float* __restrict__ feat,   // [NP,CIN] channel-last
    const float* __restrict__ out_w,  // [CIN,COUT]
    const float* __restrict__ out_b,  // [CIN]
    float* __restrict__ out)          // [B,CIN,H,W]
{
    __shared__ float lds[4 * CIN * 16];  // 64 KB
    const int lane = threadIdx.x & 31;
    const int wave = threadIdx.x >> 5;
    const int p0   = blockIdx.x * 64 + wave * 16;
    const int b    = p0 >> 14;
    const int prem = p0 & (HWp - 1);
    float* l = lds + wave * (CIN * 16);

    // Stage feat strip: global reads coalesced along c (channel-last layout).
    for (int i = lane; i < CIN * 16; i += 32) {
        int n = i >> 8, c = i & (CIN - 1);
        l[c * 16 + n] = feat[(size_t)(p0 + n) * CIN + c];
    }
    __syncthreads();

    const int half = lane >> 4;
    const int mrow = lane & 15;
    const int kp   = half * 2;
    float* ob = out + (size_t)b * COUT * HWp;

    for (int ot = 0; ot < 16; ot += 2) {
        const int o0 = ot * 16;
        v8f acc0 = {}, acc1 = {};
        for (int c0 = 0; c0 < CIN; c0 += 4) {
            v2f a0, a1, bm;
            const float* w0 = out_w + (size_t)(c0 + kp) * COUT + o0 + mrow;
            const float* w1 = w0 + COUT;
            a0.x = w0[0];  a0.y = w1[0];
            a1.x = w0[16]; a1.y = w1[16];
            bm.x = l[(c0 + kp) * 16 + mrow];
            bm.y = l[(c0 + kp + 1) * 16 + mrow];
            acc0 = __builtin_amdgcn_wmma_f32_16x16x4_f32(
                false, a0, false, bm, (short)0, acc0, false, false);
            acc1 = __builtin_amdgcn_wmma_f32_16x16x4_f32(
                false, a1, false, bm, (short)0, acc1, false, false);
        }
        // store plane-major: out[b, o, prem + n]; lanes 0-15 coalesce per r
        #pragma unroll
        for (int r = 0; r < 8; ++r) {
            const int o = o0 + 8 * half + r;
            ob[(size_t)o * HWp + prem + mrow] = acc0[r] + out_b[o];
        }
        #pragma unroll
        for (int r = 0; r < 8; ++r) {
            const int o = o0 + 16 + 8 * half + r;
            ob[(size_t)o * HWp + prem + mrow] = acc1[r] + out_b[o];
        }
    }
}

// ---------------------------------------------------------------------------
// Host launcher
// ---------------------------------------------------------------------------
extern "C" void kernel_launch(void* const* d_in, const int* in_sizes, int n_in,
                              void* d_out, int out_size, void* d_ws, size_t ws_size,
                              hipStream_t stream) {
    const float* x      = (const float*)d_in[0];
    const float* conv_w = (const float*)d_in[1];
    const float* conv_b = (const float*)d_in[2];
    const float* off_w  = (const float*)d_in[3];
    const float* off_b  = (const float*)d_in[4];
    const float* wgt_w  = (const float*)d_in[5];
    const float* wgt_b  = (const float*)d_in[6];
    const float* gamma  = (const float*)d_in[7];
    const float* beta   = (const float*)d_in[8];
    const float* rmean  = (const float*)d_in[9];
    const float* rvar   = (const float*)d_in[10];
    const float* out_w  = (const float*)d_in[11];
    const float* out_b  = (const float*)d_in[12];

    // Workspace layout (floats): proj[NP*256] | wts[NP*4] | sx[NP*4] | sy[NP*4]
    //                            | feat[NP*256]   (~262 MiB total)
    float* ws   = (float*)d_ws;
    float* proj = ws;
    float* wtsp = proj + (size_t)NP * COUT;
    float* sxp  = wtsp + (size_t)NP * KK;
    float* syp  = sxp  + (size_t)NP * KK;
    float* feat = syp  + (size_t)NP * KK;

    dfm_k1_proj<<<NP / 64, 128, 0, stream>>>(x, conv_w, conv_b, off_w, off_b,
                                             wgt_w, wgt_b, proj, wtsp, sxp, syp);
    dfm_k2_sample_bn<<<NP / 8, 256, 0, stream>>>(proj, wtsp, sxp, syp,
                                                 gamma, beta, rmean, rvar, feat);
    dfm_k3_out<<<NP / 64, 128, 0, stream>>>(feat, out_w, out_b, (float*)d_out);
}